// LossRetina4cls_8761733284316
// MI455X (gfx1250) — compile-verified
//
#include <hip/hip_runtime.h>

#define EPS_F 0.0009765625f   // 2^-10, matches torch.finfo(float16).eps
#define ALPHA_F 0.5f

typedef float v2f __attribute__((ext_vector_type(2)));
typedef float v8f __attribute__((ext_vector_type(8)));

// Full wave32 sum using the CDNA5 matrix pipe for the first fold.
// A = ones(16x4), B = {v, 0}: D[m,n] = sum_k B[k,n] = v[n] + v[n+16]
// (true for either K-row assignment of B's two VGPRs since the zero VGPR
// contributes nothing). D vgpr0 holds w[lane&15] in every lane; 4 xor
// shuffles finish the 16-wide fold. Requires EXEC == all ones.
__device__ __forceinline__ float wave_reduce_sum(float v) {
    v2f a; a[0] = 1.0f; a[1] = 1.0f;
    v2f b; b[0] = v;    b[1] = 0.0f;
    v8f c = {};
    c = __builtin_amdgcn_wmma_f32_16x16x4_f32(
        /*neg_a=*/false, a, /*neg_b=*/false, b,
        /*c_mod=*/(short)0, c, /*reuse_a=*/false, /*reuse_b=*/false);
    float w = c[0];                 // v[lane&15] + v[(lane&15)+16]
    #pragma unroll
    for (int m = 1; m < 16; m <<= 1)
        w += __shfl_xor(w, m, 32);  // fold remaining 16 distinct values
    return w;                       // full wave sum, replicated in all lanes
}

__global__ void retina_init(float* acc, int n) {
    int t = blockIdx.x * blockDim.x + threadIdx.x;
    if (t < n) acc[t] = 0.0f;
}

__global__ __launch_bounds__(256) void retina_main(
    const float* __restrict__ preg,   // [B, A, 4]
    const float* __restrict__ pcls,   // [B, A, 3]
    const int*   __restrict__ gconf,  // [B, A]
    const float* __restrict__ gcls,   // [B, A, 3]
    const float* __restrict__ gt,     // [B, A, 4]
    float* __restrict__ acc,          // [B, 3]: {cls(pos+neg), box, count}
    int A) {
    const int b   = blockIdx.y;
    const int tid = threadIdx.x;
    const long long base = (long long)b * A;

    float cls_acc = 0.0f, box_acc = 0.0f, cnt_acc = 0.0f;

    // Uniform control flow (no early return) so the WMMA below sees EXEC=~0.
    for (int i = blockIdx.x * blockDim.x + tid; i < A;
         i += blockDim.x * gridDim.x) {
        const long long n  = base + i;
        const int conf     = gconf[n];
        const float fpos   = (conf > 0) ? 1.0f : 0.0f;
        const float fneg   = (conf == 0) ? 1.0f : 0.0f;
        cnt_acc += fpos;

        // ---- focal classification loss, 3 channels ----
        const long long c3 = n * 3;
        #pragma unroll
        for (int j = 0; j < 3; ++j) {
            const float x = pcls[c3 + j];
            const float g = gcls[c3 + j];
            float p = 1.0f / (1.0f + __expf(-x));          // sigmoid
            p = fminf(fmaxf(p, EPS_F), 1.0f - EPS_F);      // clip
            const float omp  = 1.0f - p;
            const float lpos = -ALPHA_F * omp * omp * __logf(p)   * g;
            const float lneg = -ALPHA_F * p   * p   * __logf(omp) * (1.0f - g);
            cls_acc += fpos * lpos + fneg * lneg;
        }

        // ---- smooth L1 box loss (positives only) ----
        const long long c4 = n * 4;
        const float4 pr = *reinterpret_cast<const float4*>(preg + c4);
        const float4 gv = *reinterpret_cast<const float4*>(gt   + c4);
        float bl = 0.0f;
        {
            float d;
            d = fabsf(pr.x - gv.x); bl += (d < 1.0f) ? 0.5f * d * d : d - 0.5f;
            d = fabsf(pr.y - gv.y); bl += (d < 1.0f) ? 0.5f * d * d : d - 0.5f;
            d = fabsf(pr.z - gv.z); bl += (d < 1.0f) ? 0.5f * d * d : d - 0.5f;
            d = fabsf(pr.w - gv.w); bl += (d < 1.0f) ? 0.5f * d * d : d - 0.5f;
        }
        box_acc += fpos * bl;
    }

    // ---- wave reduction via WMMA + shuffles ----
    const float cls_sum = wave_reduce_sum(cls_acc);
    const float box_sum = wave_reduce_sum(box_acc);
    const float cnt_sum = wave_reduce_sum(cnt_acc);

    __shared__ float sm[8][3];
    const int wave = tid >> 5;
    const int lane = tid & 31;
    if (lane == 0) {
        sm[wave][0] = cls_sum;
        sm[wave][1] = box_sum;
        sm[wave][2] = cnt_sum;
    }
    __syncthreads();
    if (tid == 0) {
        float s0 = 0.0f, s1 = 0.0f, s2 = 0.0f;
        #pragma unroll
        for (int w = 0; w < 8; ++w) {
            s0 += sm[w][0]; s1 += sm[w][1]; s2 += sm[w][2];
        }
        atomicAdd(&acc[b * 3 + 0], s0);
        atomicAdd(&acc[b * 3 + 1], s1);
        atomicAdd(&acc[b * 3 + 2], s2);
    }
}

__global__ void retina_final(const float* __restrict__ acc,
                             float* __restrict__ out, int B) {
    const int b = threadIdx.x;
    float v = 0.0f;
    if (b < B) {
        const float cls = acc[b * 3 + 0];
        const float box = acc[b * 3 + 1];
        const float cnt = acc[b * 3 + 2];
        const float np  = fmaxf(cnt, EPS_F);
        v = (7.0f * cls + box) / np;
    }
    #pragma unroll
    for (int m = 1; m < 32; m <<= 1) v += __shfl_xor(v, m, 32);
    if (b == 0) out[0] = v / (float)B;
}

extern "C" void kernel_launch(void* const* d_in, const int* in_sizes, int n_in,
                              void* d_out, int out_size, void* d_ws, size_t ws_size,
                              hipStream_t stream) {
    (void)n_in; (void)out_size; (void)ws_size;
    const float* preg  = (const float*)d_in[0];
    const float* pcls  = (const float*)d_in[1];
    const int*   gconf = (const int*)  d_in[2];
    const float* gcls  = (const float*)d_in[3];
    const float* gt    = (const float*)d_in[4];
    float* out = (float*)d_out;
    float* acc = (float*)d_ws;          // [B][3] float accumulators

    const int B = 32;
    const int A = in_sizes[2] / B;      // gconf is [B, A]

    retina_init<<<1, 128, 0, stream>>>(acc, B * 3);

    dim3 grid(128, B);                  // 4096 blocks x 256 thr: BW-saturating
    retina_main<<<grid, 256, 0, stream>>>(preg, pcls, gconf, gcls, gt, acc, A);

    retina_final<<<1, 32, 0, stream>>>(acc, out, B);
}